// CountSketch_88819923681646
// MI455X (gfx1250) — compile-verified
//
#include <hip/hip_runtime.h>
#include <hip/hip_bf16.h>

// CountSketch on MI455X (gfx1250).
// out[n, b*512 + i_hash[b,d]] += x[n,d] * s_hash[b,d] / sqrt(B)
//
// Memory-bound (64 MB read + 128 MB write @ 23.3 TB/s ~ 8us). Strategy:
//  - workgroup = 256 threads, owns TR=8 rows x all 8 blocks
//  - LDS out-tile [8][4096] f32 (128 KB), scatter via ds_add_f32 atomics
//  - x streamed global->LDS with async DMA (global_load_async_to_lds_b128,
//    non-temporal: each element read exactly once), double buffered,
//    retired with s_wait_asynccnt
//  - result written LDS->global with global_store_async_from_lds_b128
//    (non-temporal: write-once stream; LDS tile layout == global layout)

#define D_IN    2048
#define NBLK    8
#define BS      512
#define DF      (NBLK * BS)      // 4096
#define TR      8                // rows per workgroup
#define CHUNK   256              // d-columns per chunk
#define NCHUNK  (D_IN / CHUNK)   // 8
#define NTHR    256

#define RSQRT_B 0.35355339059327373f

typedef float v4f __attribute__((ext_vector_type(4)));

// ---- CDNA5 async global<->LDS ops (ASYNCcnt) ----------------------------
__device__ __forceinline__ void async_load_b128(const float* g, const float* l) {
    unsigned           lds_a = (unsigned)(unsigned long long)l;  // low 32b of flat addr = LDS offset
    unsigned long long ga    = (unsigned long long)g;
    asm volatile("global_load_async_to_lds_b128 %0, %1, off th:TH_LOAD_NT"
                 :: "v"(lds_a), "v"(ga) : "memory");
}

__device__ __forceinline__ void async_store_b128(float* g, const float* l) {
    unsigned           lds_a = (unsigned)(unsigned long long)l;
    unsigned long long ga    = (unsigned long long)g;
    asm volatile("global_store_async_from_lds_b128 %0, %1, off th:TH_STORE_NT"
                 :: "v"(ga), "v"(lds_a) : "memory");
}

__device__ __forceinline__ void wait_async_le2() {
    asm volatile("s_wait_asynccnt 2" ::: "memory");
}
__device__ __forceinline__ void wait_async_0() {
    asm volatile("s_wait_asynccnt 0" ::: "memory");
}

// Issue the async DMA for one x chunk: rows [row0,row0+TR) x dims [c*CHUNK, ...).
// Flattened chunk = TR*CHUNK = 2048 floats = 512 b128 transfers; 256 threads -> 2 each.
__device__ __forceinline__ void issue_chunk(const float* __restrict__ x,
                                            int row0, int c, float* stage, int t) {
#pragma unroll
    for (int k = 0; k < 2; ++k) {
        int j   = t + k * NTHR;          // float4 index in [0,512)
        int r   = j >> 6;                // j / (CHUNK/4)
        int dd4 = j & 63;                // j % (CHUNK/4)
        const float* g = x + (size_t)(row0 + r) * D_IN + c * CHUNK + dd4 * 4;
        const float* l = stage + r * CHUNK + dd4 * 4;
        async_load_b128(g, l);
    }
}

__global__ __launch_bounds__(NTHR)
void countsketch_kernel(const float* __restrict__ x,
                        const float* __restrict__ s_hash,
                        const int*   __restrict__ i_hash,
                        float*       __restrict__ out) {
    extern __shared__ float lds[];
    float* out_tile = lds;                 // TR*DF floats   (128 KB)
    float* x_stage  = lds + TR * DF;       // 2 * TR*CHUNK   (16 KB)

    const int t    = threadIdx.x;
    const int row0 = blockIdx.x * TR;

    // Zero the out-tile: TR*DF = 32768 floats = 8192 float4; exactly 32 per thread.
#pragma unroll 4
    for (int k = 0; k < (TR * DF / 4) / NTHR; ++k) {
        v4f z = {0.f, 0.f, 0.f, 0.f};
        ((v4f*)out_tile)[k * NTHR + t] = z;
    }

    // Prefetch chunk 0 (touches x_stage only; zero-fill is on out_tile -> no barrier yet).
    issue_chunk(x, row0, 0, x_stage, t);

    for (int c = 0; c < NCHUNK; ++c) {
        float* xcur = x_stage + (c & 1) * (TR * CHUNK);

        if (c + 1 < NCHUNK) {
            // Prefetch next chunk into the other buffer (that buffer was last
            // read in iter c-1; the trailing barrier there protects it).
            issue_chunk(x, row0, c + 1, x_stage + ((c + 1) & 1) * (TR * CHUNK), t);
            wait_async_le2();   // async loads retire in order: chunk c is now in LDS
        } else {
            wait_async_0();     // last chunk: drain everything
        }
        __syncthreads();        // whole tile (all waves' DMA) visible; also covers zero-fill

        // Per-thread hash state for column d = c*CHUNK + t (reused across TR rows).
        const int d = c * CHUNK + t;
        float sb[NBLK];
        int   off[NBLK];
#pragma unroll
        for (int b = 0; b < NBLK; ++b) {
            sb[b]  = s_hash[b * D_IN + d] * RSQRT_B;
            off[b] = b * BS + i_hash[b * D_IN + d];
        }

        // Scatter-accumulate: 8 rows x 8 blocks of ds_add_f32 per thread.
        // LDS bank = bucket mod 64 (random) -> ~1.3x serialization expected.
#pragma unroll
        for (int r = 0; r < TR; ++r) {
            float xv   = xcur[r * CHUNK + t];     // ds_load_b32, conflict-free
            int   base = r * DF;
#pragma unroll
            for (int b = 0; b < NBLK; ++b) {
                atomicAdd(&out_tile[base + off[b]], xv * sb[b]);
            }
        }
        __syncthreads();   // all waves done reading xcur before it is overwritten
    }

    // Out-tile is the exact contiguous image of rows [row0, row0+TR) of out.
    // Stream it out with async LDS->global DMA (no VGPR round trip), NT.
    float* gbase = out + (size_t)row0 * DF;
#pragma unroll 4
    for (int k = 0; k < (TR * DF / 4) / NTHR; ++k) {
        int i = k * NTHR + t;
        async_store_b128(gbase + i * 4, out_tile + i * 4);
    }
    wait_async_0();        // LDS must stay live until DMA completes
}

extern "C" void kernel_launch(void* const* d_in, const int* in_sizes, int n_in,
                              void* d_out, int out_size, void* d_ws, size_t ws_size,
                              hipStream_t stream) {
    const float* x      = (const float*)d_in[0];
    const float* s_hash = (const float*)d_in[1];
    const int*   i_hash = (const int*)d_in[2];
    float*       out    = (float*)d_out;

    const int N = in_sizes[0] / D_IN;           // 8192
    const int grid = N / TR;                    // 1024 workgroups
    const size_t shmem = (size_t)(TR * DF + 2 * TR * CHUNK) * sizeof(float); // 147456 B

    countsketch_kernel<<<dim3(grid), dim3(NTHR), shmem, stream>>>(x, s_hash, i_hash, out);
}